// GLASELinkPrediction_41120016892382
// MI455X (gfx1250) — compile-verified
//
#include <hip/hip_runtime.h>

typedef __attribute__((ext_vector_type(2))) float v2f;
typedef __attribute__((ext_vector_type(8))) float v8f;
typedef __attribute__((ext_vector_type(4))) unsigned int v4u;
typedef __attribute__((ext_vector_type(8))) int v8i;
typedef __attribute__((ext_vector_type(4))) int v4i;

#define TPB 256

#if defined(__HIP_DEVICE_COMPILE__) && defined(__has_builtin)
#if __has_builtin(__builtin_amdgcn_tensor_load_to_lds) && __has_builtin(__builtin_amdgcn_s_wait_tensorcnt)
#define USE_TDM 1
#endif
#endif
#ifndef USE_TDM
#define USE_TDM 0
#endif

// ---------------- degree / normalization ----------------
__global__ void k_deg_count(const int* __restrict__ dst, float* __restrict__ deg, int E) {
  int e = blockIdx.x * blockDim.x + threadIdx.x;
  if (e < E) atomicAdd(&deg[dst[e]], 1.0f);
}

__global__ void k_dinv(float* __restrict__ deg, int N) {
  int i = blockIdx.x * blockDim.x + threadIdx.x;
  if (i < N) deg[i] = rsqrtf(deg[i] + 1.0f);
}

// ---------------- GD block ----------------
// per node: xq = x * diag(Q); y1 = (x @ W1) * diag(Q); y2 = (x @ W2) * diag(Q)
__global__ void k_gd_prep(const float* __restrict__ x, const float* __restrict__ W1,
                          const float* __restrict__ W2, const float* __restrict__ Q,
                          float* __restrict__ xq, float* __restrict__ y1,
                          float* __restrict__ y2, int N) {
  int i = blockIdx.x * blockDim.x + threadIdx.x;
  if (i >= N) return;
  const float4* xp = (const float4*)(x + (size_t)i * 8);
  float4 lo = xp[0], hi = xp[1];
  float xv[8] = {lo.x, lo.y, lo.z, lo.w, hi.x, hi.y, hi.z, hi.w};
#pragma unroll
  for (int j = 0; j < 8; ++j) {
    float s = Q[j * 8 + j];          // Q is diag(+-1)
    float a1 = 0.f, a2 = 0.f;
#pragma unroll
    for (int k = 0; k < 8; ++k) {
      a1 += xv[k] * W1[k * 8 + j];
      a2 += xv[k] * W2[k * 8 + j];
    }
    xq[(size_t)i * 8 + j] = xv[j] * s;
    y1[(size_t)i * 8 + j] = a1 * s;
    y2[(size_t)i * 8 + j] = a2 * s;
  }
}

// t1: acc[dst] += y1[src]  over A edges
__global__ void k_gd_edgeA(const int* __restrict__ src, const int* __restrict__ dst,
                           const float* __restrict__ y1, float* __restrict__ acc, int E) {
  int e = blockIdx.x * blockDim.x + threadIdx.x;
  if (e >= E) return;
  int s = src[e], d = dst[e];
  const float4* yp = (const float4*)(y1 + (size_t)s * 8);
  float4 a = yp[0], b = yp[1];
  float* ap = acc + (size_t)d * 8;
  atomicAdd(ap + 0, a.x); atomicAdd(ap + 1, a.y);
  atomicAdd(ap + 2, a.z); atomicAdd(ap + 3, a.w);
  atomicAdd(ap + 4, b.x); atomicAdd(ap + 5, b.y);
  atomicAdd(ap + 6, b.z); atomicAdd(ap + 7, b.w);
}

// t2: w = <xq[ms], x[md]>;  acc[ms] -= w * y2[md]   over M edges
__global__ void k_gd_edgeM(const int* __restrict__ msrc, const int* __restrict__ mdst,
                           const float* __restrict__ xq, const float* __restrict__ x,
                           const float* __restrict__ y2, float* __restrict__ acc, int EM) {
  int e = blockIdx.x * blockDim.x + threadIdx.x;
  if (e >= EM) return;
  int s = msrc[e], d = mdst[e];
  const float4* qp = (const float4*)(xq + (size_t)s * 8);
  const float4* xp = (const float4*)(x + (size_t)d * 8);
  float4 q0 = qp[0], q1 = qp[1], x0 = xp[0], x1 = xp[1];
  float w = q0.x * x0.x + q0.y * x0.y + q0.z * x0.z + q0.w * x0.w +
            q1.x * x1.x + q1.y * x1.y + q1.z * x1.z + q1.w * x1.w;
  const float4* yp = (const float4*)(y2 + (size_t)d * 8);
  float4 a = yp[0], b = yp[1];
  float* ap = acc + (size_t)s * 8;
  atomicAdd(ap + 0, -w * a.x); atomicAdd(ap + 1, -w * a.y);
  atomicAdd(ap + 2, -w * a.z); atomicAdd(ap + 3, -w * a.w);
  atomicAdd(ap + 4, -w * b.x); atomicAdd(ap + 5, -w * b.y);
  atomicAdd(ap + 6, -w * b.z); atomicAdd(ap + 7, -w * b.w);
}

__global__ void k_axpy(float* __restrict__ x, const float* __restrict__ acc, int n) {
  int i = blockIdx.x * blockDim.x + threadIdx.x;
  if (i < n) x[i] += acc[i];
}

// ---------------- f32 WMMA GEMM with LDS-staged B ----------------
// C[M x (NT*16)] = [A0 | A1] @ B.  A0: k in [0,K0), A1: k in [K0,K).
// Block = 8 waves; each wave computes a 16-row x NT*16-col panel so the A
// fragment is loaded once per K-chunk and reused across NT WMMAs.
// B (K x NC, contiguous) is staged in LDS once per block — via the Tensor
// Data Mover when available (tensor_load_to_lds + s_wait_tensorcnt), else a
// cooperative copy. All waves read B fragments from LDS (ds_load).
template <int NT, int KMAX>
__global__ void __launch_bounds__(TPB)
k_gemm_wmma_lds(const float* __restrict__ A0, int lda0, int K0,
                const float* __restrict__ A1, int lda1,
                const float* __restrict__ B,
                float* __restrict__ C, int M, int K) {
  constexpr int NC = NT * 16;
  __shared__ float Blds[KMAX * NC];

#if USE_TDM
  if ((threadIdx.x >> 5) == 0) {
    // D# descriptor (CDNA5 ISA): 2D tile == whole B.
    unsigned long long ga = (unsigned long long)(const void*)B;
    unsigned int ldsb = (unsigned int)(unsigned long long)(void*)&Blds[0]; // flat addr low 32 = LDS byte offset
    v4u g0;
    g0.x = 1u;                                            // count=1, user descriptor
    g0.y = ldsb;                                          // lds_addr
    g0.z = (unsigned int)ga;                              // global_addr[31:0]
    g0.w = (unsigned int)((ga >> 32) & 0x1FFFFFFu) | (2u << 30); // addr[56:32] | type=2
    unsigned int td0 = (unsigned int)NC;                  // tensor_dim0 (elements per row)
    unsigned int td1 = (unsigned int)K;                   // tensor_dim1 (rows)
    v8i g1;
    g1[0] = (int)(2u << 16);                              // data_size = 4 bytes
    g1[1] = (int)((td0 & 0xFFFFu) << 16);                 // tensor_dim0 lo16 @ bits[63:48]
    g1[2] = (int)((td0 >> 16) | ((td1 & 0xFFFFu) << 16)); // dim0 hi16 | dim1 lo16
    g1[3] = (int)((td1 >> 16) | ((unsigned)NC << 16));    // dim1 hi16 | tile_dim0 = NC
    g1[4] = (int)(td1 & 0xFFFFu);                         // tile_dim1 = K rows (tile_dim2 = 0)
    g1[5] = (int)NC;                                      // tensor_dim0_stride lo32
    g1[6] = 0;                                            // stride hi16 | dim1_stride lo16
    g1[7] = 0;
    v4i gz = {0, 0, 0, 0};
#if __clang_major__ >= 23
    v8i gz8 = {0, 0, 0, 0, 0, 0, 0, 0};
    __builtin_amdgcn_tensor_load_to_lds(g0, g1, gz, gz, gz8, 0);
#else
    __builtin_amdgcn_tensor_load_to_lds(g0, g1, gz, gz, 0);
#endif
    __builtin_amdgcn_s_wait_tensorcnt(0);
  }
#else
  for (int i = threadIdx.x; i < K * NC; i += blockDim.x) Blds[i] = B[i];
#endif
  __syncthreads();

  int mt = blockIdx.x * (TPB / 32) + (threadIdx.x >> 5);
  int mtiles = (M + 15) >> 4;
  if (mt >= mtiles) return;                 // wave-uniform exit; EXEC all-1s for WMMA
  int lane = threadIdx.x & 31;
  int half = lane >> 4, lid = lane & 15;
  int row0 = mt * 16;
  int arow = row0 + lid; if (arow >= M) arow = M - 1;  // clamped load, masked store

  v8f acc[NT];
#pragma unroll
  for (int t = 0; t < NT; ++t) acc[t] = {};

  for (int k = 0; k < K; k += 4) {
    int kk = k + half * 2;                  // ISA 16x4 A layout: lanes 0-15 -> K=0,1; 16-31 -> K=2,3
    v2f a;
    if (kk < K0) {
      const float* p = A0 + (size_t)arow * lda0 + kk;
      a.x = p[0]; a.y = p[1];
    } else {
      const float* p = A1 + (size_t)arow * lda1 + (kk - K0);
      a.x = p[0]; a.y = p[1];
    }
#pragma unroll
    for (int t = 0; t < NT; ++t) {
      v2f b;
      b.x = Blds[kk * NC + t * 16 + lid];
      b.y = Blds[(kk + 1) * NC + t * 16 + lid];
      acc[t] = __builtin_amdgcn_wmma_f32_16x16x4_f32(false, a, false, b, (short)0,
                                                     acc[t], false, false);
    }
  }

#pragma unroll
  for (int t = 0; t < NT; ++t) {
#pragma unroll
    for (int r = 0; r < 8; ++r) {
      int row = row0 + r + half * 8;
      if (row < M) C[(size_t)row * NC + t * 16 + lid] = acc[t][r];
    }
  }
}

// ---------------- GCN aggregation ----------------
// out[n,f] = h[n,f]*dinv[n]^2 + bias[f]   (self-loop term + bias)
__global__ void k_init_bias(const float* __restrict__ h, const float* __restrict__ dinv,
                            const float* __restrict__ bias, float* __restrict__ out,
                            int N, int F) {
  int idx = blockIdx.x * blockDim.x + threadIdx.x;
  if (idx >= N * F) return;
  int n = idx / F, f = idx - n * F;
  float di = dinv[n];
  out[idx] = h[idx] * di * di + bias[f];
}

// out[dst] += h[src] * dinv[src]*dinv[dst], 4 floats per thread
__global__ void k_gcn_agg(const float* __restrict__ h, const int* __restrict__ src,
                          const int* __restrict__ dst, const float* __restrict__ dinv,
                          float* __restrict__ out, int E, int F) {
  int chunks = F >> 2;
  int t = blockIdx.x * blockDim.x + threadIdx.x;
  if (t >= E * chunks) return;
  int e = t / chunks, c = t - e * chunks;
  int s = src[e], d = dst[e];
  float w = dinv[s] * dinv[d];
  float4 v = *(const float4*)(h + (size_t)s * F + c * 4);
  float* op = out + (size_t)d * F + c * 4;
  atomicAdd(op + 0, v.x * w); atomicAdd(op + 1, v.y * w);
  atomicAdd(op + 2, v.z * w); atomicAdd(op + 3, v.w * w);
}

__global__ void k_relu(float* __restrict__ x, int n) {
  int i = blockIdx.x * blockDim.x + threadIdx.x;
  if (i < n) x[i] = fmaxf(x[i], 0.0f);
}

// ---------------- decoder ----------------
__global__ void k_decode(const float* __restrict__ z, const int* __restrict__ a,
                         const int* __restrict__ b, float* __restrict__ logits,
                         int EL, int F) {
  int e = blockIdx.x * blockDim.x + threadIdx.x;
  if (e >= EL) return;
  const float4* za = (const float4*)(z + (size_t)a[e] * F);
  const float4* zb = (const float4*)(z + (size_t)b[e] * F);
  float acc = 0.f;
  for (int c = 0; c < (F >> 2); ++c) {
    float4 u = za[c], v = zb[c];
    acc += u.x * v.x + u.y * v.y + u.z * v.z + u.w * v.w;
  }
  logits[e] = acc;
}

static inline int nblk(long long n) { return (int)((n + TPB - 1) / TPB); }

extern "C" void kernel_launch(void* const* d_in, const int* in_sizes, int n_in,
                              void* d_out, int out_size, void* d_ws, size_t ws_size,
                              hipStream_t stream) {
  const float* x_feat = (const float*)d_in[0];
  const float* x_init = (const float*)d_in[1];
  const float* Q      = (const float*)d_in[2];
  const float* Wgd1   = (const float*)d_in[3];
  const float* Wgd2   = (const float*)d_in[4];
  const float* Wc1    = (const float*)d_in[5];
  const float* bc1    = (const float*)d_in[6];
  const float* Wc2    = (const float*)d_in[7];
  const float* bc2    = (const float*)d_in[8];
  const int*   eA     = (const int*)d_in[9];   // edge_index (2,E)
  const int*   eM     = (const int*)d_in[11];  // mask_edges (2,EM)
  const int*   eL     = (const int*)d_in[12];  // edge_label_index (2,EL)

  const int EMB  = 8;
  const int N    = in_sizes[1] / EMB;
  const int FEAT = in_sizes[0] / N;
  const int STEPS = in_sizes[3] / (EMB * EMB);
  const int HID  = in_sizes[5] / (FEAT + EMB);
  const int OUT  = in_sizes[7] / HID;
  const int E    = in_sizes[9] / 2;
  const int EM   = in_sizes[11] / 2;
  const int EL   = in_sizes[12] / 2;

  const int* srcA = eA;        const int* dstA = eA + E;
  const int* srcM = eM;        const int* dstM = eM + EM;
  const int* srcL = eL;        const int* dstL = eL + EL;

  // workspace layout (floats)
  float* ws   = (float*)d_ws;
  float* dinv = ws;                           // N
  float* xA   = dinv + N;                     // N*8  (current embedding)
  float* xq   = xA + (size_t)N * 8;           // N*8
  float* y1   = xq + (size_t)N * 8;           // N*8
  float* y2   = y1 + (size_t)N * 8;           // N*8
  float* acc  = y2 + (size_t)N * 8;           // N*8
  float* h    = acc + (size_t)N * 8;          // N*HID (GEMM out; reused as h2 N*OUT)
  float* hout = h + (size_t)N * HID;          // N*HID (post-agg layer1)

  float* logits   = (float*)d_out;            // EL
  float* z        = logits + EL;              // N*OUT
  float* xemb_out = z + (size_t)N * OUT;      // N*8

  // --- degree normalization (depends only on edge_index) ---
  hipMemsetAsync(dinv, 0, (size_t)N * sizeof(float), stream);
  k_deg_count<<<nblk(E), TPB, 0, stream>>>(dstA, dinv, E);
  k_dinv<<<nblk(N), TPB, 0, stream>>>(dinv, N);

  // --- GD block unroll ---
  hipMemcpyAsync(xA, x_init, (size_t)N * 8 * sizeof(float), hipMemcpyDeviceToDevice, stream);
  for (int t = 0; t < STEPS; ++t) {
    k_gd_prep<<<nblk(N), TPB, 0, stream>>>(xA, Wgd1 + (size_t)t * 64, Wgd2 + (size_t)t * 64,
                                           Q, xq, y1, y2, N);
    hipMemsetAsync(acc, 0, (size_t)N * 8 * sizeof(float), stream);
    k_gd_edgeA<<<nblk(E), TPB, 0, stream>>>(srcA, dstA, y1, acc, E);
    k_gd_edgeM<<<nblk(EM), TPB, 0, stream>>>(srcM, dstM, xq, xA, y2, acc, EM);
    k_axpy<<<nblk((long long)N * 8), TPB, 0, stream>>>(xA, acc, N * 8);
  }

  int mtiles = (N + 15) / 16;
  int gblocks = (mtiles + (TPB / 32) - 1) / (TPB / 32);

  // --- GCN layer 1: h = [x_feat | x_emb] @ Wc1 (f32 WMMA, B in LDS, HID=128 -> NT=8) ---
  k_gemm_wmma_lds<8, 136><<<gblocks, TPB, 0, stream>>>(
      x_feat, FEAT, FEAT, xA, EMB, Wc1, h, N, FEAT + EMB);
  k_init_bias<<<nblk((long long)N * HID), TPB, 0, stream>>>(h, dinv, bc1, hout, N, HID);
  k_gcn_agg<<<nblk((long long)E * (HID / 4)), TPB, 0, stream>>>(h, srcA, dstA, dinv, hout, E, HID);
  k_relu<<<nblk((long long)N * HID), TPB, 0, stream>>>(hout, N * HID);

  // --- GCN layer 2: h2 = hout @ Wc2 (OUT=64 -> NT=4) ---
  k_gemm_wmma_lds<4, 128><<<gblocks, TPB, 0, stream>>>(
      hout, HID, HID, hout, HID, Wc2, h, N, HID);
  k_init_bias<<<nblk((long long)N * OUT), TPB, 0, stream>>>(h, dinv, bc2, z, N, OUT);
  k_gcn_agg<<<nblk((long long)E * (OUT / 4)), TPB, 0, stream>>>(h, srcA, dstA, dinv, z, E, OUT);

  // --- decoder + x_emb output ---
  k_decode<<<nblk(EL), TPB, 0, stream>>>(z, srcL, dstL, logits, EL, OUT);
  hipMemcpyAsync(xemb_out, xA, (size_t)N * 8 * sizeof(float), hipMemcpyDeviceToDevice, stream);
}